// NestedFormula_36077725287013
// MI455X (gfx1250) — compile-verified
//
#include <hip/hip_runtime.h>
#include <stdint.h>

// NestedFormula: DEPTH=4, V=4, B=131072.
// n1=125 level-1 nodes, n2=25, n3=5, n4=1, 125 depth-0 constants.
//
// Strategy: one lane per batch element (wave32), depth-first fully-unrolled
// tree evaluation in registers. All ~1373 parameter floats are staged once
// per workgroup into LDS with CDNA5 async global->LDS loads (ASYNCcnt),
// interleaved as (lam,pow) float2 pairs so the hot loop does one ds_load_b64
// broadcast per term.

#define THREADS 256

// LDS byte layout:
//   pairs (float2 = {lam, pow}):
//     L1: pair [   0, 500)  -> bytes [    0, 4000)
//     L2: pair [ 500, 600)  -> bytes [ 4000, 4800)
//     L3: pair [ 600, 620)  -> bytes [ 4800, 4960)
//     L4: pair [ 620, 624)  -> bytes [ 4960, 4992)
//   lam0: 125 floats after the pairs.

__device__ __forceinline__ void async_ld_b32(uint32_t lds_byte, const float* base,
                                             uint32_t byte_off) {
  // CDNA5 async copy: copies a dword from base + byte_off into LDS at lds_byte; tracked by ASYNCcnt.
  asm volatile("global_load_async_to_lds_b32 %0, %1, %2"
               :: "v"(lds_byte), "v"(byte_off), "s"(base)
               : "memory");
}

__global__ __launch_bounds__(THREADS) void nested_formula_kernel(
    const float* __restrict__ x,
    const float* __restrict__ lam0,
    const float* __restrict__ lam1, const float* __restrict__ pw1,
    const float* __restrict__ lam2, const float* __restrict__ pw2,
    const float* __restrict__ lam3, const float* __restrict__ pw3,
    const float* __restrict__ lam4, const float* __restrict__ pw4,
    float* __restrict__ out, int batch)
{
  __shared__ float2 sPairs[624];
  __shared__ float  sLam0[128];

  const int tid = threadIdx.x;
  const int b   = blockIdx.x * THREADS + tid;

  // Issue the per-lane x load early so it overlaps the parameter staging.
  float4 xv = make_float4(1.f, 1.f, 1.f, 1.f);
  if (b < batch) xv = reinterpret_cast<const float4*>(x)[b];

  // Generic pointers to LDS carry the LDS byte offset in their low 32 bits.
  const uint32_t pairsBase = (uint32_t)(uintptr_t)(void*)&sPairs[0];
  const uint32_t lam0Base  = (uint32_t)(uintptr_t)(void*)&sLam0[0];

  // ---- stage parameters into LDS via async global->LDS copies ----
  for (int i = tid; i < 500; i += THREADS) {
    async_ld_b32(pairsBase + (uint32_t)i * 8u,      lam1, (uint32_t)i * 4u);
    async_ld_b32(pairsBase + (uint32_t)i * 8u + 4u, pw1,  (uint32_t)i * 4u);
  }
  if (tid < 125)
    async_ld_b32(lam0Base + (uint32_t)tid * 4u, lam0, (uint32_t)tid * 4u);
  if (tid < 100) {
    async_ld_b32(pairsBase + 4000u + (uint32_t)tid * 8u,      lam2, (uint32_t)tid * 4u);
    async_ld_b32(pairsBase + 4000u + (uint32_t)tid * 8u + 4u, pw2,  (uint32_t)tid * 4u);
  }
  if (tid < 20) {
    async_ld_b32(pairsBase + 4800u + (uint32_t)tid * 8u,      lam3, (uint32_t)tid * 4u);
    async_ld_b32(pairsBase + 4800u + (uint32_t)tid * 8u + 4u, pw3,  (uint32_t)tid * 4u);
  }
  if (tid < 4) {
    async_ld_b32(pairsBase + 4960u + (uint32_t)tid * 8u,      lam4, (uint32_t)tid * 4u);
    async_ld_b32(pairsBase + 4960u + (uint32_t)tid * 8u + 4u, pw4,  (uint32_t)tid * 4u);
  }

  asm volatile("s_wait_asynccnt 0" ::: "memory");
  __syncthreads();

  if (b >= batch) return;

  // x in (0.5, 1.5): x^p = exp2(p * log2 x). 4 logs, 624 exp2 per element.
  float lx[4];
  lx[0] = __builtin_amdgcn_logf(xv.x);
  lx[1] = __builtin_amdgcn_logf(xv.y);
  lx[2] = __builtin_amdgcn_logf(xv.z);
  lx[3] = __builtin_amdgcn_logf(xv.w);

  // Depth-first evaluation; fully unrolled so every array index is a
  // compile-time constant and all state lives in VGPRs.
  float acc4 = 0.f, last4 = 0.f;
#pragma unroll
  for (int j3 = 0; j3 < 5; ++j3) {
    float acc3 = 0.f, last3 = 0.f;
#pragma unroll
    for (int j2 = 0; j2 < 5; ++j2) {
      const int n2 = j3 * 5 + j2;
      float acc2 = 0.f, last2 = 0.f;
#pragma unroll
      for (int j1 = 0; j1 < 5; ++j1) {
        const int n1 = n2 * 5 + j1;
        float f1 = sLam0[n1];
#pragma unroll
        for (int v = 0; v < 4; ++v) {
          const float2 lp = sPairs[n1 * 4 + v];
          f1 = fmaf(lp.x, __builtin_amdgcn_exp2f(lp.y * lx[v]), f1);
        }
        if (j1 == 4) {
          last2 = f1;
        } else {
          const float2 lp = sPairs[500 + n2 * 4 + j1];
          acc2 = fmaf(lp.x * __builtin_amdgcn_exp2f(lp.y * lx[j1]), f1, acc2);
        }
      }
      const float f2 = acc2 + last2;
      if (j2 == 4) {
        last3 = f2;
      } else {
        const float2 lp = sPairs[600 + j3 * 4 + j2];
        acc3 = fmaf(lp.x * __builtin_amdgcn_exp2f(lp.y * lx[j2]), f2, acc3);
      }
    }
    const float f3 = acc3 + last3;
    if (j3 == 4) {
      last4 = f3;
    } else {
      const float2 lp = sPairs[620 + j3];
      acc4 = fmaf(lp.x * __builtin_amdgcn_exp2f(lp.y * lx[j3]), f3, acc4);
    }
  }

  out[b] = acc4 + last4;
}

extern "C" void kernel_launch(void* const* d_in, const int* in_sizes, int n_in,
                              void* d_out, int out_size, void* d_ws, size_t ws_size,
                              hipStream_t stream) {
  (void)n_in; (void)out_size; (void)d_ws; (void)ws_size;
  const float* x    = (const float*)d_in[0];
  const float* lam0 = (const float*)d_in[1];
  const float* lam1 = (const float*)d_in[2];
  const float* pw1  = (const float*)d_in[3];
  const float* lam2 = (const float*)d_in[4];
  const float* pw2  = (const float*)d_in[5];
  const float* lam3 = (const float*)d_in[6];
  const float* pw3  = (const float*)d_in[7];
  const float* lam4 = (const float*)d_in[8];
  const float* pw4  = (const float*)d_in[9];
  float* out = (float*)d_out;

  const int batch  = in_sizes[0] / 4;           // x is (B, 4)
  const int blocks = (batch + THREADS - 1) / THREADS;

  nested_formula_kernel<<<blocks, THREADS, 0, stream>>>(
      x, lam0, lam1, pw1, lam2, pw2, lam3, pw3, lam4, pw4, out, batch);
}